// SRSChannelEstimator_46832323396424
// MI455X (gfx1250) — compile-verified
//
#include <hip/hip_runtime.h>
#include <math.h>

typedef __attribute__((ext_vector_type(2))) float v2f;
typedef __attribute__((ext_vector_type(8))) float v8f;

#define SEQ_L 3264
#define KTC   12
#define NPORT 32
#define BLK   12
#define NBLK  272              // SEQ_L / BLK
#define MROWS 8704             // NPORT * NBLK
#define NWIN  21               // DMAX-DMIN+1
#define DMIN  (-10)
#define LOCC  4
#define NAVG  816              // SEQ_L / LOCC
#define RHO   0.9f
#define TWO_PI_OVER_L (6.28318530717958647692f / (float)SEQ_L)

// workspace layout (float offsets)
#define OFF_NOISE   0
#define OFF_PEAK    1                              // 32 ints
#define OFF_B       64                             // 12x16 padded F^T
#define OFF_HAVG_RE 256
#define OFF_HAVG_IM (OFF_HAVG_RE + NPORT*NAVG)     // 26368
#define OFF_RES_RE  (OFF_HAVG_IM + NPORT*NAVG)     // 52480
#define OFF_RES_IM  (OFF_RES_RE + SEQ_L)           // 55744
#define OFF_HWR_RE  (OFF_RES_IM + SEQ_L)           // 59008
#define OFF_HWR_IM  (OFF_HWR_RE + NPORT*SEQ_L)     // 163456
#define OFF_HMMSE   (OFF_HWR_IM + NPORT*SEQ_L)     // 267904, interleaved complex

// ---------------- kernel 1: noise power + per-port CIR peak search ----------------
__global__ void k_noise_peaks(const float* __restrict__ lr, const float* __restrict__ li,
                              const int* __restrict__ cs, float* __restrict__ ws) {
    __shared__ float sRe[256];
    __shared__ float sIm[256];
    const int t = threadIdx.x;
    if (blockIdx.x == 0) {
        // noise_power = mean(|ls[1:]-ls[:-1]|^2)/2
        float acc = 0.f;
        for (int n = t; n < SEQ_L - 1; n += 256) {
            float dr = lr[n + 1] - lr[n], di = li[n + 1] - li[n];
            acc += dr * dr + di * di;
        }
        sRe[t] = acc; __syncthreads();
        for (int s = 128; s > 0; s >>= 1) { if (t < s) sRe[t] += sRe[t + s]; __syncthreads(); }
        if (t == 0) ws[OFF_NOISE] = sRe[0] / (2.0f * (float)(SEQ_L - 1));
    } else {
        const int p = blockIdx.x - 1;
        int c = cs[p];
        int ideal = (((KTC - c) % KTC) + KTC) % KTC * (SEQ_L / KTC);
        float best = -1.f; int bestn = SEQ_L;
        for (int j = 0; j < NWIN; ++j) {
            int n  = ideal + DMIN + j;
            int nm = ((n % SEQ_L) + SEQ_L) % SEQ_L;
            float ar = 0.f, ai = 0.f;
            for (int k = t; k < SEQ_L; k += 256) {
                int ph = (k * nm) % SEQ_L;                 // < 2^31, exact
                float th = (float)ph * TWO_PI_OVER_L;
                float s, cth; __sincosf(th, &s, &cth);
                ar += lr[k] * cth - li[k] * s;             // x * e^{+j th}  (ifft sign)
                ai += lr[k] * s   + li[k] * cth;
            }
            sRe[t] = ar; sIm[t] = ai; __syncthreads();
            for (int sd = 128; sd > 0; sd >>= 1) {
                if (t < sd) { sRe[t] += sRe[t + sd]; sIm[t] += sIm[t + sd]; }
                __syncthreads();
            }
            if (t == 0) {
                float m2 = sRe[0] * sRe[0] + sIm[0] * sIm[0];
                // jnp.argmax: smallest absolute index wins ties
                if (m2 > best || (m2 == best && nm < bestn)) { best = m2; bestn = nm; }
            }
            __syncthreads();
        }
        if (t == 0) ((int*)ws)[OFF_PEAK + p] = bestn;
    }
}

// ---------------- kernel 2: F^T = A^{-1} C  (12x12, one thread) ----------------
__global__ void k_mmse_filter(float* __restrict__ ws) {
    if (threadIdx.x != 0 || blockIdx.x != 0) return;
    float A[BLK][BLK], X[BLK][BLK], rp[BLK];
    const float np_ = ws[OFF_NOISE];
    rp[0] = 1.f;
    for (int i = 1; i < BLK; ++i) rp[i] = rp[i - 1] * RHO;
    for (int i = 0; i < BLK; ++i)
        for (int j = 0; j < BLK; ++j) {
            float cij = rp[i > j ? i - j : j - i];
            A[i][j] = cij + (i == j ? np_ : 0.f);
            X[i][j] = cij;                      // RHS = C
        }
    // Gauss-Jordan (A is SPD): X := A^{-1} C == F^T
    for (int col = 0; col < BLK; ++col) {
        float inv = 1.0f / A[col][col];
        for (int j = 0; j < BLK; ++j) { A[col][j] *= inv; X[col][j] *= inv; }
        for (int r = 0; r < BLK; ++r) if (r != col) {
            float f = A[r][col];
            for (int j = 0; j < BLK; ++j) { A[r][j] -= f * A[col][j]; X[r][j] -= f * X[col][j]; }
        }
    }
    // padded B matrix for WMMA: B[k][j] = F[j][k], j=12..15 zero
    for (int k = 0; k < BLK; ++k)
        for (int j = 0; j < 16; ++j)
            ws[OFF_B + k * 16 + j] = (j < BLK) ? X[k][j] : 0.f;
}

// ---------------- kernel 3: derotate + OCC average-by-4 ----------------
__global__ void k_havg(const float* __restrict__ lr, const float* __restrict__ li,
                       float* __restrict__ ws) {
    int idx = blockIdx.x * blockDim.x + threadIdx.x;
    if (idx >= NPORT * NAVG) return;
    int p = idx / NAVG, i = idx % NAVG;
    int peak = ((const int*)ws)[OFF_PEAK + p];
    float sr = 0.f, si = 0.f;
    for (int q = 0; q < LOCC; ++q) {
        int n  = i * LOCC + q;
        int ph = (peak * n) % SEQ_L;
        float th = (float)ph * TWO_PI_OVER_L;
        float s, c; __sincosf(th, &s, &c);
        sr += lr[n] * c - li[n] * s;
        si += lr[n] * s + li[n] * c;
    }
    ws[OFF_HAVG_RE + idx] = sr * 0.25f;
    ws[OFF_HAVG_IM + idx] = si * 0.25f;
}

__device__ __forceinline__ void interp_port(const float* ws, int p, int i0, float tf,
                                            float& hr, float& hi) {
    int base = p * NAVG + i0;
    hr = ws[OFF_HAVG_RE + base];
    hi = ws[OFF_HAVG_IM + base];
    if (tf > 0.f) {
        hr += tf * (ws[OFF_HAVG_RE + base + 1] - hr);
        hi += tf * (ws[OFF_HAVG_IM + base + 1] - hi);
    }
}

// ---------------- kernel 4: residual = ls - sum_p h_interp * conj(phasor) ----------------
__global__ void k_residual(const float* __restrict__ lr, const float* __restrict__ li,
                           float* __restrict__ ws) {
    int n = blockIdx.x * blockDim.x + threadIdx.x;
    if (n >= SEQ_L) return;
    float pos = ((float)n - 1.5f) * 0.25f;       // jnp.interp sample grid 1.5 + 4*i
    int i0; float tf;
    if (pos <= 0.f)             { i0 = 0;        tf = 0.f; }
    else if (pos >= NAVG - 1)   { i0 = NAVG - 1; tf = 0.f; }
    else                        { i0 = (int)pos; tf = pos - (float)i0; }
    float accr = 0.f, acci = 0.f;
    for (int p = 0; p < NPORT; ++p) {
        float hr, hi; interp_port(ws, p, i0, tf, hr, hi);
        int peak = ((const int*)ws)[OFF_PEAK + p];
        int ph = (peak * n) % SEQ_L;
        float th = (float)ph * TWO_PI_OVER_L;
        float s, c; __sincosf(th, &s, &c);
        accr += hr * c + hi * s;                 // * conj(phasor)
        acci += hi * c - hr * s;
    }
    ws[OFF_RES_RE + n] = lr[n] - accr;
    ws[OFF_RES_IM + n] = li[n] - acci;
}

// ---------------- kernel 5: h_wr = h_interp + residual * phasor (planar, GEMM layout) ----------------
__global__ void k_hwr(float* __restrict__ ws) {
    int idx = blockIdx.x * blockDim.x + threadIdx.x;
    if (idx >= NPORT * SEQ_L) return;
    int p = idx / SEQ_L, n = idx % SEQ_L;
    float pos = ((float)n - 1.5f) * 0.25f;
    int i0; float tf;
    if (pos <= 0.f)             { i0 = 0;        tf = 0.f; }
    else if (pos >= NAVG - 1)   { i0 = NAVG - 1; tf = 0.f; }
    else                        { i0 = (int)pos; tf = pos - (float)i0; }
    float hr, hi; interp_port(ws, p, i0, tf, hr, hi);
    int peak = ((const int*)ws)[OFF_PEAK + p];
    int ph = (peak * n) % SEQ_L;
    float th = (float)ph * TWO_PI_OVER_L;
    float s, c; __sincosf(th, &s, &c);
    float rr = ws[OFF_RES_RE + n], ri = ws[OFF_RES_IM + n];
    ws[OFF_HWR_RE + idx] = hr + rr * c - ri * s;
    ws[OFF_HWR_IM + idx] = hi + rr * s + ri * c;
}

// ---------------- kernel 6: block-MMSE GEMM via V_WMMA_F32_16X16X4_F32 ----------------
// out[m, j] = sum_k hwr[m, k] * F[j, k]   (m = 0..8703 rows of 12, B = F^T padded to 16 cols)
__global__ void k_mmse_wmma(float* __restrict__ ws) {
    const int wave = threadIdx.x >> 5;
    const int tile = blockIdx.x * (blockDim.x >> 5) + wave;   // 544 tiles of 16 rows
    const int lane = threadIdx.x & 31;
    const int half = lane >> 4, lm = lane & 15;
    const float* are = ws + OFF_HWR_RE;
    const float* aim = ws + OFF_HWR_IM;
    const float* B   = ws + OFF_B;
    const int rowBase = tile * 16;
    v8f cre = {}; v8f cim = {};
#pragma unroll
    for (int c = 0; c < 3; ++c) {                 // K = 12 in chunks of 4
        int k0   = 4 * c + 2 * half;              // A/B fragment: v holds K = k0 + v
        int aIdx = (rowBase + lm) * 12 + k0;      // even index -> 8B aligned
        v2f aR = *(const v2f*)(are + aIdx);
        v2f aI = *(const v2f*)(aim + aIdx);
        v2f b;
        b.x = B[k0 * 16 + lm];
        b.y = B[(k0 + 1) * 16 + lm];
        cre = __builtin_amdgcn_wmma_f32_16x16x4_f32(false, aR, false, b, (short)0, cre, false, false);
        cim = __builtin_amdgcn_wmma_f32_16x16x4_f32(false, aI, false, b, (short)0, cim, false, false);
    }
    if (lm < BLK) {
        float2* out = (float2*)(ws + OFF_HMMSE);
#pragma unroll
        for (int r = 0; r < 8; ++r) {             // D: VGPR r, lane -> M = r + 8*half, N = lm
            int row = rowBase + r + 8 * half;
            out[row * 12 + lm] = make_float2(cre[r], cim[r]);
        }
    }
}

// ---------------- kernel 7: tile [32,L] -> [1024,L], float4 streaming ----------------
__global__ void k_tile(const float* __restrict__ ws, float* __restrict__ out) {
    const long total4 = (long)NPORT * NPORT * SEQ_L / 2;      // complex pairs
    long idx = (long)blockIdx.x * blockDim.x + threadIdx.x;
    if (idx >= total4) return;
    const float4* src = (const float4*)(ws + OFF_HMMSE);
    long srcIdx = idx % ((long)NPORT * SEQ_L / 2);            // 52224
    ((float4*)out)[idx] = src[srcIdx];
}

extern "C" void kernel_launch(void* const* d_in, const int* in_sizes, int n_in,
                              void* d_out, int out_size, void* d_ws, size_t ws_size,
                              hipStream_t stream) {
    const float* lr = (const float*)d_in[0];
    const float* li = (const float*)d_in[1];
    const int*   cs = (const int*)d_in[2];
    float* ws  = (float*)d_ws;
    float* out = (float*)d_out;

    k_noise_peaks<<<NPORT + 1, 256, 0, stream>>>(lr, li, cs, ws);
    k_mmse_filter<<<1, 32, 0, stream>>>(ws);
    k_havg<<<(NPORT * NAVG + 255) / 256, 256, 0, stream>>>(lr, li, ws);
    k_residual<<<(SEQ_L + 255) / 256, 256, 0, stream>>>(lr, li, ws);
    k_hwr<<<(NPORT * SEQ_L + 255) / 256, 256, 0, stream>>>(ws);
    k_mmse_wmma<<<MROWS / 16 / 4, 128, 0, stream>>>(ws);      // 136 blocks x 4 waves
    long total4 = (long)NPORT * NPORT * SEQ_L / 2;
    k_tile<<<(unsigned)((total4 + 255) / 256), 256, 0, stream>>>(ws, out);
}